// SpikeFP32Divider_23407571764116
// MI455X (gfx1250) — compile-verified
//
#include <hip/hip_runtime.h>
#include <stdint.h>

// ---------------------------------------------------------------------------
// SpikeFP32Divider: pulse(32 f32 bits) -> fp32 divide -> pulse(32 f32 bits)
// Memory-bound streaming kernel using CDNA5 async global<->LDS data movers.
// ---------------------------------------------------------------------------

#define WPB        4                    // waves per block (wave32)
#define TILE_ELEMS 32                   // fp32 words per tile (one per lane)
#define ROWF       36                   // padded LDS row stride in floats (144B, 16B aligned)
#define TILE_FLOATS (TILE_ELEMS * 32)   // 1024 floats = 4KB per tile
#define REGION_F   (TILE_ELEMS * ROWF)  // 1152 floats per LDS region
#define WAVE_LDS_F (3 * REGION_F)       // A, B, O regions per wave

// b128 async builtins take int-vector(16B) pointers in AS(1)/AS(3).
typedef int v4i __attribute__((vector_size(16)));
typedef __attribute__((address_space(1))) v4i* gv4_p;
typedef __attribute__((address_space(3))) v4i* lv4_p;

// Flat shared pointers carry the LDS byte offset in their low 32 bits
// (ISA 10.2, LDS aperture). Integer round-trip cast
// gives us a well-formed address-space(3) pointer.
__device__ __forceinline__ lv4_p to_lds(const void* p) {
  return (lv4_p)(uint32_t)(uintptr_t)p;
}
__device__ __forceinline__ gv4_p to_glb(const void* p) {
  return (gv4_p)(uintptr_t)p;
}

// ---- CDNA5 async global->LDS / LDS->global (ASYNCcnt-tracked) -------------
__device__ __forceinline__ void async_load_b128(const void* gsrc, void* ldst) {
#if __has_builtin(__builtin_amdgcn_global_load_async_to_lds_b128)
  __builtin_amdgcn_global_load_async_to_lds_b128(to_glb(gsrc), to_lds(ldst), 0, 0);
#else
  uint32_t lo = (uint32_t)(uintptr_t)ldst;
  uint64_t ga = (uint64_t)(uintptr_t)gsrc;
  asm volatile("global_load_async_to_lds_b128 %0, %1, off"
               :: "v"(lo), "v"(ga) : "memory");
#endif
}

__device__ __forceinline__ void async_store_b128(void* gdst, const void* lsrc) {
#if __has_builtin(__builtin_amdgcn_global_store_async_from_lds_b128)
  __builtin_amdgcn_global_store_async_from_lds_b128(to_glb(gdst), to_lds(lsrc), 0, 0);
#else
  uint32_t lo = (uint32_t)(uintptr_t)lsrc;
  uint64_t ga = (uint64_t)(uintptr_t)gdst;
  asm volatile("global_store_async_from_lds_b128 %0, %1, off"
               :: "v"(ga), "v"(lo) : "memory");
#endif
}

// Hardware waits as inline asm with "memory" clobbers: guarantees both the
// s_wait_* SOPP emission AND a compiler-level fence so LDS reads/writes are
// not scheduled across the hardware synchronization point.
__device__ __forceinline__ void wait_async0() {
  asm volatile("s_wait_asynccnt 0x0" ::: "memory");
}
__device__ __forceinline__ void wait_ds0() {
  asm volatile("s_wait_dscnt 0x0" ::: "memory");
}

__global__ void __launch_bounds__(WPB * 32)
spike_fp32_div_kernel(const float* __restrict__ A,
                      const float* __restrict__ B,
                      float* __restrict__ O,
                      int nElem) {
  __shared__ float lds[WPB * WAVE_LDS_F] __attribute__((aligned(16)));

  const int lane      = threadIdx.x & 31;
  const int waveInBlk = threadIdx.x >> 5;
  const int waveId    = blockIdx.x * WPB + waveInBlk;
  const int nWaves    = gridDim.x * WPB;

  float* ldsA = lds + waveInBlk * WAVE_LDS_F;
  float* ldsB = ldsA + REGION_F;
  float* ldsO = ldsB + REGION_F;

  const int fullTiles  = nElem >> 5;
  const int totalTiles = (nElem + 31) >> 5;

  for (int tile = waveId; tile < totalTiles; tile += nWaves) {
    if (tile < fullTiles) {
      const size_t gbase = (size_t)tile * TILE_FLOATS;  // float index of tile

      // ---- stage 2 x 4KB tiles into LDS: contiguous 16B chunks per lane ---
      // chunk i of 256 -> LDS row i/8, slot i%8 (row stride padded to 144B)
#pragma unroll
      for (int k = 0; k < 8; ++k) {
        const int i    = k * 32 + lane;
        const int loff = (i >> 3) * ROWF + (i & 7) * 4;
        async_load_b128(A + gbase + (size_t)i * 4, ldsA + loff);
        async_load_b128(B + gbase + (size_t)i * 4, ldsB + loff);
      }
      wait_async0();  // also drains the previous iteration's async store

      // ---- lane l owns element l: read its (transposed) row from LDS -----
      const float4* ra = (const float4*)(ldsA + lane * ROWF);
      const float4* rb = (const float4*)(ldsB + lane * ROWF);
      uint32_t ua = 0u, ub = 0u;
#pragma unroll
      for (int k = 0; k < 8; ++k) {
        const float4 va = ra[k];
        const float4 vb = rb[k];
        // pulses are exact 0.0f/1.0f -> v_cvt_u32_f32 + shift/or chains
        ua = (ua << 4) | ((uint32_t)va.x << 3) | ((uint32_t)va.y << 2) |
             ((uint32_t)va.z << 1) | (uint32_t)va.w;
        ub = (ub << 4) | ((uint32_t)vb.x << 3) | ((uint32_t)vb.y << 2) |
             ((uint32_t)vb.z << 1) | (uint32_t)vb.w;
      }

      // IEEE-754 RNE fp32 division (default HIP codegen: div_scale/fmas/fixup)
      const float q  = __uint_as_float(ua) / __uint_as_float(ub);
      const uint32_t uq = __float_as_uint(q);

      // ---- unpack quotient bits into this lane's LDS output row ----------
      float4* ro = (float4*)(ldsO + lane * ROWF);
#pragma unroll
      for (int k = 0; k < 8; ++k) {
        const int base = 31 - 4 * k;
        float4 v;
        v.x = (float)((uq >> base) & 1u);
        v.y = (float)((uq >> (base - 1)) & 1u);
        v.z = (float)((uq >> (base - 2)) & 1u);
        v.w = (float)((uq >> (base - 3)) & 1u);
        ro[k] = v;
      }
      wait_ds0();  // LDS row writes visible before async engine reads them

      // ---- async store output tile, contiguous in global memory ----------
#pragma unroll
      for (int k = 0; k < 8; ++k) {
        const int i    = k * 32 + lane;
        const int loff = (i >> 3) * ROWF + (i & 7) * 4;
        async_store_b128(O + gbase + (size_t)i * 4, ldsO + loff);
      }
    } else {
      // -------- tail (< 32 elements): simple per-lane direct path ---------
      const int e = tile * 32 + lane;
      if (e < nElem) {
        const size_t base = (size_t)e * 32;
        uint32_t ua = 0u, ub = 0u;
        for (int j = 0; j < 32; ++j) {
          ua = (ua << 1) | (uint32_t)A[base + j];
          ub = (ub << 1) | (uint32_t)B[base + j];
        }
        const float q  = __uint_as_float(ua) / __uint_as_float(ub);
        const uint32_t uq = __float_as_uint(q);
        for (int j = 0; j < 32; ++j) {
          O[base + j] = (float)((uq >> (31 - j)) & 1u);
        }
      }
    }
  }
  // S_ENDPGM performs an implicit wait-idle, draining outstanding async stores.
}

extern "C" void kernel_launch(void* const* d_in, const int* in_sizes, int n_in,
                              void* d_out, int out_size, void* d_ws, size_t ws_size,
                              hipStream_t stream) {
  const float* A = (const float*)d_in[0];
  const float* B = (const float*)d_in[1];
  float* O = (float*)d_out;

  const long long totalFloats = (long long)in_sizes[0];
  const int nElem = (int)(totalFloats / 32);          // fp32 words
  const int totalTiles = (nElem + 31) / 32;

  int blocks = (totalTiles + WPB - 1) / WPB;
  if (blocks > 4096) blocks = 4096;                   // grid-stride beyond this
  if (blocks < 1) blocks = 1;

  spike_fp32_div_kernel<<<dim3(blocks), dim3(WPB * 32), 0, stream>>>(A, B, O, nElem);
}